// AudioDDCMCodebook_2044404433535
// MI455X (gfx1250) — compile-verified
//
#include <hip/hip_runtime.h>
#include <hip/hip_bf16.h>

typedef float v2f __attribute__((ext_vector_type(2)));
typedef float v4f __attribute__((ext_vector_type(4)));
typedef float v8f __attribute__((ext_vector_type(8)));

#define B_SZ 64
#define K_SZ 4096
#define D_SZ 32000

// ---------------------------------------------------------------------------
// Kernel 1: distance tiles via V_WMMA_F32_16X16X4_F32.
// Grid: K_SZ/16 = 256 blocks, 128 threads (4 waves).
// Wave w computes the 16x16 tile  dist[16w..16w+15][16*blockIdx.x .. +15].
//
// WMMA f32 16x16x4 lane layouts (ISA 7.12.2):
//   A (16x4):  lane l holds row M=l%16;  v0 = A[M][2*(l/16)+0], v1 = A[M][2*(l/16)+1]
//   B (4x16):  lane l holds col N=l%16;  v0 = B[2*(l/16)+0][N], v1 = B[2*(l/16)+1][N]
//   C/D:       vgpr r, lane l -> M = r + 8*(l/16), N = l%16
//
// K-permutation trick: the reduction over D is order-agnostic, so each lane
// does ONE global_load_b128 at (d + 4*half) and feeds TWO wmma steps:
//   wmma #1 uses .xy  -> covers k = {d, d+1, d+4, d+5}
//   wmma #2 uses .zw  -> covers k = {d+2, d+3, d+6, d+7}
// A and B use the same permutation, so the dot product is exact. This halves
// vmem instruction count vs b64 feeds and keeps WMMA operands on naturally
// even-aligned VGPR sub-pairs (no repacking movs).
// ---------------------------------------------------------------------------
__global__ __launch_bounds__(128) void vq_dist_kernel(
    const float* __restrict__ latent,
    const float* __restrict__ codebook,
    float* __restrict__ dist)
{
    const int lane = threadIdx.x & 31;
    const int wave = threadIdx.x >> 5;   // 0..3 -> latent row block
    const int col  = lane & 15;          // A row / B col selector
    const int half = lane >> 4;          // K-group selector
    const int n_base = blockIdx.x * 16;

    const float* __restrict__ Ap = latent   + (size_t)(wave * 16 + col) * D_SZ + 4 * half;
    const float* __restrict__ Bp = codebook + (size_t)(n_base   + col) * D_SZ + 4 * half;

    v8f acc = {};
    v4f x2a = {};   // elementwise |latent|^2 partials (keeps pk_fma on load regs)
    v4f c2a = {};   // elementwise |codebook|^2 partials

    // D_SZ = 32000 = 1000 * 32 ; 4 x (b128 load pair + 2 wmma) per iter.
    for (int d = 0; d < D_SZ; d += 32) {
        // stay ~8KB ahead on the codebook stream (HBM-bound operand)
        if (d + 2048 < D_SZ) __builtin_prefetch(Bp + d + 2048, 0, 1);
#pragma unroll
        for (int u = 0; u < 4; ++u) {
            v4f a = *(const v4f*)(Ap + d + u * 8);
            v4f b = *(const v4f*)(Bp + d + u * 8);
            x2a += a * a;            // v_pk_fma_f32 on the loaded pairs
            c2a += b * b;
            // (neg_a, A, neg_b, B, c_mod, C, reuse_a, reuse_b)
            acc = __builtin_amdgcn_wmma_f32_16x16x4_f32(
                false, (v2f)a.xy, false, (v2f)b.xy, (short)0, acc, false, false);
            acc = __builtin_amdgcn_wmma_f32_16x16x4_f32(
                false, (v2f)a.zw, false, (v2f)b.zw, (short)0, acc, false, false);
        }
    }

    float x2p = (x2a.x + x2a.y) + (x2a.z + x2a.w);
    float c2p = (c2a.x + c2a.y) + (c2a.z + c2a.w);

    // Combine the two lane-halves: lanes l and l+16 together cover all of D.
    float x2full = x2p + __shfl_xor(x2p, 16);   // |x|^2 of latent row (wave*16+col)
    float c2full = c2p + __shfl_xor(c2p, 16);   // |c|^2 of codebook row (n_base+col)

#pragma unroll
    for (int r = 0; r < 8; ++r) {
        const int m = r + 8 * half;             // M index of acc[r] in this lane
        const float xm = __shfl(x2full, m);     // lane m holds |x|^2 of row m
        float d2 = xm + c2full - 2.0f * acc[r];
        float dd = sqrtf(fmaxf(d2, 0.0f));
        dist[(size_t)(wave * 16 + m) * K_SZ + n_base + col] = dd;
    }
}

// ---------------------------------------------------------------------------
// Kernel 2a: new_usage = usage_in (then 2b atomically adds the histogram)
// ---------------------------------------------------------------------------
__global__ __launch_bounds__(256) void usage_init_kernel(
    const float* __restrict__ usage_in, float* __restrict__ usage_out)
{
    int i = blockIdx.x * blockDim.x + threadIdx.x;
    if (i < K_SZ) usage_out[i] = usage_in[i];
}

// ---------------------------------------------------------------------------
// Kernel 2b: per-row argmin over K=4096, write index (as float), min dist,
// and atomic-add the usage histogram. 64 blocks x 128 threads.
// ---------------------------------------------------------------------------
__global__ __launch_bounds__(128) void argmin_kernel(
    const float* __restrict__ dist,
    float* __restrict__ out_idx,
    float* __restrict__ out_mind,
    float* __restrict__ usage_out)
{
    __shared__ float svals[128];
    __shared__ int   sidx[128];
    const int row = blockIdx.x;
    const int tid = threadIdx.x;

    const float* __restrict__ drow = dist + (size_t)row * K_SZ;
    float best = 3.402823466e+38f;
    int   bi   = 0;
    for (int j = tid; j < K_SZ; j += 128) {
        float v = drow[j];
        if (v < best) { best = v; bi = j; }   // j increases per-thread: first-min kept
    }
    svals[tid] = best;
    sidx[tid]  = bi;
    __syncthreads();
#pragma unroll
    for (int s = 64; s > 0; s >>= 1) {
        if (tid < s) {
            float ov = svals[tid + s];
            int   oi = sidx[tid + s];
            if (ov < svals[tid] || (ov == svals[tid] && oi < sidx[tid])) {
                svals[tid] = ov; sidx[tid] = oi;
            }
        }
        __syncthreads();
    }
    if (tid == 0) {
        out_idx[row]  = (float)sidx[0];
        out_mind[row] = svals[0];
        atomicAdd(&usage_out[sidx[0]], 1.0f);   // global_atomic_add_f32
    }
}

// ---------------------------------------------------------------------------
// Kernel 3: quantized[b] = codebook[idx[b]]. Grid (125, 64) x 256 threads;
// 125*256 = 32000 contiguous floats per row. Overwrites the scratch distance
// region, which has already been consumed by argmin (stream-ordered).
// ---------------------------------------------------------------------------
__global__ __launch_bounds__(256) void gather_kernel(
    const float* __restrict__ codebook,
    const float* __restrict__ idx_f,
    float* __restrict__ quant)
{
    const int b = blockIdx.y;
    const int i = blockIdx.x * 256 + threadIdx.x;
    const int k = (int)idx_f[b];
    quant[(size_t)b * D_SZ + i] = codebook[(size_t)k * D_SZ + i];
}

// ---------------------------------------------------------------------------
extern "C" void kernel_launch(void* const* d_in, const int* in_sizes, int n_in,
                              void* d_out, int out_size, void* d_ws, size_t ws_size,
                              hipStream_t stream)
{
    (void)in_sizes; (void)n_in; (void)out_size; (void)d_ws; (void)ws_size;

    const float* latent   = (const float*)d_in[0];   // [64, 8,250,16]  f32
    const float* codebook = (const float*)d_in[1];   // [4096, 8,250,16] f32
    const float* usage_in = (const float*)d_in[2];   // [4096] f32

    float* out       = (float*)d_out;
    float* quant     = out;                               // 64*32000 floats
    float* out_idx   = out + (size_t)B_SZ * D_SZ;         // 64
    float* out_mind  = out_idx + B_SZ;                    // 64
    float* usage_out = out_mind + B_SZ;                   // 4096

    // Reuse the quantized region as scratch for the 64x4096 distance matrix
    // (262144 floats << 2,048,000 floats); gather_kernel overwrites it last.
    float* dist = quant;

    vq_dist_kernel   <<<dim3(K_SZ / 16),  dim3(128), 0, stream>>>(latent, codebook, dist);
    usage_init_kernel<<<dim3(K_SZ / 256), dim3(256), 0, stream>>>(usage_in, usage_out);
    argmin_kernel    <<<dim3(B_SZ),       dim3(128), 0, stream>>>(dist, out_idx, out_mind, usage_out);
    gather_kernel    <<<dim3(125, B_SZ),  dim3(256), 0, stream>>>(codebook, out_idx, quant);
}